// AutoRegerting_2954937500106
// MI455X (gfx1250) — compile-verified
//
#include <hip/hip_runtime.h>
#include <hip/hip_bf16.h>
#include <math.h>

// Problem constants (match reference)
#define Vv   32000
#define Ee   512
#define Hh   1024
#define Bb   16
#define Tt   256
#define H3   (3 * Hh)
#define EPSL 1e-5f

typedef float v2f __attribute__((ext_vector_type(2)));
typedef float v8f __attribute__((ext_vector_type(8)));

// ---------------------------------------------------------------------------
// Generic WMMA f32 GEMM:  C[M,N] = A[M,K] @ W[N,K]^T + bias[N]
// One wave computes MT stacked 16x16 output tiles (reuses each B fragment MT
// times -> cuts W traffic MT x for the big logits GEMM).
// Uses V_WMMA_F32_16X16X4_F32 (wave32, exact f32).
// A 16x4 frag: lanes 0-15 hold {K0,K1} of row M=lane, lanes 16-31 hold {K2,K3}
// B 4x16 frag: lanes 0-15 hold {K0,K1} of col N=lane, lanes 16-31 hold {K2,K3}
// C/D: VGPR r holds row (r + 8*laneHi), col = lane&15.
// K loop is software-pipelined with PREFETCH DISTANCE 2 (two rotating
// fragment buffers, chunk loop unrolled by 2 so indices stay static): the
// fragments consumed at chunk c were loaded at chunk c-2, so steady-state
// waits tolerate the newest chunk's loads in flight instead of draining.
// ---------------------------------------------------------------------------
template <int MT>
__global__ __launch_bounds__(256) void wmma_gemm_bias_kernel(
    const float* __restrict__ A,
    const float* __restrict__ W,
    const float* __restrict__ bias,
    float* __restrict__ C,
    int M, int N, int K) {
  const int wavesPerBlock = blockDim.x >> 5;
  const int tileId = blockIdx.x * wavesPerBlock + (threadIdx.x >> 5);
  const int lane = threadIdx.x & 31;
  const int nTilesN = N >> 4;
  const int nTilesM = (M >> 4) / MT;
  if (tileId >= nTilesM * nTilesN) return;   // uniform per-wave: EXEC stays all-1s
  const int tn = tileId % nTilesN;
  const int tm = tileId / nTilesN;
  const int n0 = tn << 4;
  const int m0 = tm * (16 * MT);

  const int hi  = (lane >= 16) ? 1 : 0;
  const int l15 = lane & 15;
  const int kk0 = hi << 1;                   // per-lane K sub-offset

  v8f acc[MT];
#pragma unroll
  for (int i = 0; i < MT; ++i)
#pragma unroll
    for (int r = 0; r < 8; ++r) acc[i][r] = 0.0f;

  const float* wcol = W + (size_t)(n0 + l15) * K;  // row n of W == col n of B
  const float* arow[MT];
#pragma unroll
  for (int i = 0; i < MT; ++i)
    arow[i] = A + (size_t)(m0 + i * 16 + l15) * K;

  auto loadFrag = [&](v2f& bF, v2f (&aF)[MT], int chunk) {
    const int kk = (chunk << 2) + kk0;
    bF.x = wcol[kk];
    bF.y = wcol[kk + 1];
#pragma unroll
    for (int i = 0; i < MT; ++i) {
      aF[i].x = arow[i][kk];
      aF[i].y = arow[i][kk + 1];
    }
  };
  auto mma = [&](const v2f& bF, const v2f (&aF)[MT]) {
#pragma unroll
    for (int i = 0; i < MT; ++i)
      acc[i] = __builtin_amdgcn_wmma_f32_16x16x4_f32(
          false, aF[i], false, bF, (short)0, acc[i], false, false);
  };

  const int nc = K >> 2;                     // number of K=4 chunks (>= 2, even)
  v2f b0, b1;
  v2f a0[MT], a1[MT];
  loadFrag(b0, a0, 0);
  loadFrag(b1, a1, 1);

  int c = 0;
  for (; c + 3 < nc; c += 2) {
    mma(b0, a0);
    loadFrag(b0, a0, c + 2);                 // refill slot0 with chunk c+2
    mma(b1, a1);
    loadFrag(b1, a1, c + 3);                 // refill slot1 with chunk c+3
  }
  if (c + 2 < nc) {                          // 3 chunks left
    mma(b0, a0);
    loadFrag(b0, a0, c + 2);
    mma(b1, a1);
    mma(b0, a0);
  } else {                                   // 2 chunks left
    mma(b0, a0);
    mma(b1, a1);
  }

  const float bn = bias ? bias[n0 + l15] : 0.0f;
#pragma unroll
  for (int i = 0; i < MT; ++i) {
    const int rbase = m0 + i * 16 + (hi ? 8 : 0);
#pragma unroll
    for (int r = 0; r < 8; ++r)
      C[(size_t)(rbase + r) * N + n0 + l15] = acc[i][r] + bn;
  }
}

// ---------------------------------------------------------------------------
// Embedding gather: Xe[b*T+t, :] = embd[input[b*T+t], :]
// ---------------------------------------------------------------------------
__global__ void embed_kernel(const int* __restrict__ idx,
                             const float* __restrict__ embd,
                             float* __restrict__ Xe) {
  const int row = blockIdx.x;            // b*T + t
  const int tid = threadIdx.x;           // 256 threads, E=512 -> 2/thread
  const int tok = idx[row];
  const float* src = embd + (size_t)tok * Ee;
  float* dst = Xe + (size_t)row * Ee;
  dst[tid] = src[tid];
  dst[tid + 256] = src[tid + 256];
}

__global__ void zero_kernel(float* __restrict__ p, int n) {
  int i = blockIdx.x * blockDim.x + threadIdx.x;
  if (i < n) p[i] = 0.0f;
}

// ---------------------------------------------------------------------------
// GRU gates + LayerNorm. One block per batch row b (16 blocks, 256 threads,
// 4 H-elements per thread). Carried state is the post-LN hidden (matches ref).
//   gi row for b:  gi + b*giRowStride           (already offset to step t)
//   gh row for b:  gh + b*3H
//   optionally also writes the post-LN value to hseq + b*hseqRowStride
// ---------------------------------------------------------------------------
__global__ void gru_gates_ln_kernel(const float* __restrict__ gi, long giRowStride,
                                    const float* __restrict__ gh,
                                    const float* __restrict__ hprev,
                                    const float* __restrict__ lnw,
                                    const float* __restrict__ lnb,
                                    float* __restrict__ hout,
                                    float* __restrict__ hseq, long hseqRowStride) {
  const int b = blockIdx.x;
  const int tid = threadIdx.x;
  const float* giR = gi + (size_t)b * giRowStride;
  const float* ghR = gh + (size_t)b * H3;
  const float* hR = hprev + (size_t)b * Hh;

  float vals[4];
  float sum = 0.0f, sumsq = 0.0f;
#pragma unroll
  for (int i = 0; i < 4; ++i) {
    const int hx = tid + i * 256;
    const float ir = giR[hx], iz = giR[Hh + hx], inn = giR[2 * Hh + hx];
    const float hr = ghR[hx], hz = ghR[Hh + hx], hn = ghR[2 * Hh + hx];
    const float r = 1.0f / (1.0f + __expf(-(ir + hr)));
    const float z = 1.0f / (1.0f + __expf(-(iz + hz)));
    const float n = tanhf(inn + r * hn);
    const float hv = (1.0f - z) * n + z * hR[hx];
    vals[i] = hv;
    sum += hv;
    sumsq += hv * hv;
  }
  // wave32 reduce, then cross-wave via LDS
  for (int off = 16; off > 0; off >>= 1) {
    sum += __shfl_xor(sum, off);
    sumsq += __shfl_xor(sumsq, off);
  }
  __shared__ float s1[8], s2[8];
  const int wid = tid >> 5;
  if ((tid & 31) == 0) { s1[wid] = sum; s2[wid] = sumsq; }
  __syncthreads();
  if (wid == 0) {
    float a = (tid < 8) ? s1[tid] : 0.0f;
    float c = (tid < 8) ? s2[tid] : 0.0f;
    for (int off = 4; off > 0; off >>= 1) {
      a += __shfl_xor(a, off);
      c += __shfl_xor(c, off);
    }
    if (tid == 0) { s1[0] = a; s2[0] = c; }
  }
  __syncthreads();
  const float mean = s1[0] * (1.0f / Hh);
  const float var = s2[0] * (1.0f / Hh) - mean * mean;
  const float inv = rsqrtf(var + EPSL);

  float* hOutR = hout + (size_t)b * Hh;
  float* hSeqR = hseq ? (hseq + (size_t)b * hseqRowStride) : nullptr;
#pragma unroll
  for (int i = 0; i < 4; ++i) {
    const int hx = tid + i * 256;
    const float y = (vals[i] - mean) * inv * lnw[hx] + lnb[hx];
    hOutR[hx] = y;
    if (hSeqR) hSeqR[hx] = y;
  }
}

// ---------------------------------------------------------------------------
// LeakyReLU(0.01) + LayerNorm, in place, one block per row of H=1024.
// ---------------------------------------------------------------------------
__global__ void lrelu_ln_kernel(float* __restrict__ X,
                                const float* __restrict__ lnw,
                                const float* __restrict__ lnb) {
  const int row = blockIdx.x;
  const int tid = threadIdx.x;
  float* xR = X + (size_t)row * Hh;

  float vals[4];
  float sum = 0.0f, sumsq = 0.0f;
#pragma unroll
  for (int i = 0; i < 4; ++i) {
    const int hx = tid + i * 256;
    float x = xR[hx];
    x = (x < 0.0f) ? 0.01f * x : x;
    vals[i] = x;
    sum += x;
    sumsq += x * x;
  }
  for (int off = 16; off > 0; off >>= 1) {
    sum += __shfl_xor(sum, off);
    sumsq += __shfl_xor(sumsq, off);
  }
  __shared__ float s1[8], s2[8];
  const int wid = tid >> 5;
  if ((tid & 31) == 0) { s1[wid] = sum; s2[wid] = sumsq; }
  __syncthreads();
  if (wid == 0) {
    float a = (tid < 8) ? s1[tid] : 0.0f;
    float c = (tid < 8) ? s2[tid] : 0.0f;
    for (int off = 4; off > 0; off >>= 1) {
      a += __shfl_xor(a, off);
      c += __shfl_xor(c, off);
    }
    if (tid == 0) { s1[0] = a; s2[0] = c; }
  }
  __syncthreads();
  const float mean = s1[0] * (1.0f / Hh);
  const float var = s2[0] * (1.0f / Hh) - mean * mean;
  const float inv = rsqrtf(var + EPSL);
#pragma unroll
  for (int i = 0; i < 4; ++i) {
    const int hx = tid + i * 256;
    xR[hx] = (vals[i] - mean) * inv * lnw[hx] + lnb[hx];
  }
}

// ---------------------------------------------------------------------------
template <int MT>
static inline void launch_gemm(const float* A, const float* W, const float* bias,
                               float* C, int M, int N, int K, hipStream_t stream) {
  const int tiles = (M / (16 * MT)) * (N / 16);
  const int wavesPerBlock = 8;  // 256 threads = 8 wave32
  const int blocks = (tiles + wavesPerBlock - 1) / wavesPerBlock;
  wmma_gemm_bias_kernel<MT><<<blocks, wavesPerBlock * 32, 0, stream>>>(A, W, bias, C, M, N, K);
}

extern "C" void kernel_launch(void* const* d_in, const int* in_sizes, int n_in,
                              void* d_out, int out_size, void* d_ws, size_t ws_size,
                              hipStream_t stream) {
  const int*   input = (const int*)d_in[0];
  const float* embd  = (const float*)d_in[1];
  const float* Wih0  = (const float*)d_in[2];
  const float* Whh0  = (const float*)d_in[3];
  const float* bih0  = (const float*)d_in[4];
  const float* bhh0  = (const float*)d_in[5];
  const float* ln0w  = (const float*)d_in[6];
  const float* ln0b  = (const float*)d_in[7];
  const float* Wih1  = (const float*)d_in[8];
  const float* Whh1  = (const float*)d_in[9];
  const float* bih1  = (const float*)d_in[10];
  const float* bhh1  = (const float*)d_in[11];
  const float* ln1w  = (const float*)d_in[12];
  const float* ln1b  = (const float*)d_in[13];
  const float* W1    = (const float*)d_in[14];
  const float* b1    = (const float*)d_in[15];
  const float* ln2w  = (const float*)d_in[16];
  const float* ln2b  = (const float*)d_in[17];
  const float* W2    = (const float*)d_in[18];
  const float* b2    = (const float*)d_in[19];
  float* out = (float*)d_out;

  // ---- workspace carving (~93 MB of f32) --------------------------------
  float* ws = (float*)d_ws;
  float* Xe    = ws;                               // [4096, 512]
  float* GI0   = Xe + (size_t)Bb * Tt * Ee;        // [4096, 3072]
  float* h0    = GI0 + (size_t)Bb * Tt * H3;       // [16, 1024]
  float* h1    = h0 + (size_t)Bb * Hh;             // [16, 1024]
  float* gh0b  = h1 + (size_t)Bb * Hh;             // [16, 3072]
  float* gi1b  = gh0b + (size_t)Bb * H3;           // [16, 3072]
  float* gh1b  = gi1b + (size_t)Bb * H3;           // [16, 3072]
  float* H1seq = gh1b + (size_t)Bb * H3;           // [4096, 1024]
  float* Act   = H1seq + (size_t)Bb * Tt * Hh;     // [4096, 1024]

  // h0/h1 are contiguous: zero both (harness does not re-poison between calls)
  zero_kernel<<<(2 * Bb * Hh + 255) / 256, 256, 0, stream>>>(h0, 2 * Bb * Hh);

  // ---- phase 1 (parallel): embed + input-side GEMM for all timesteps ----
  embed_kernel<<<Bb * Tt, 256, 0, stream>>>(input, embd, Xe);
  launch_gemm<4>(Xe, Wih0, bih0, GI0, Bb * Tt, H3, Ee, stream);   // [4096,3072]

  // ---- phase 2 (sequential over T): recurrent GEMMs + gates + LN --------
  for (int t = 0; t < Tt; ++t) {
    // layer 0: gh0 = h0 @ Whh0^T + bhh0   (M=16, N=3072, K=1024)
    launch_gemm<1>(h0, Whh0, bhh0, gh0b, Bb, H3, Hh, stream);
    gru_gates_ln_kernel<<<Bb, 256, 0, stream>>>(
        GI0 + (size_t)t * H3, (long)Tt * H3, gh0b, h0, ln0w, ln0b, h0,
        nullptr, 0);
    // layer 1: gi1 = h0 @ Wih1^T + bih1 ; gh1 = h1 @ Whh1^T + bhh1
    launch_gemm<1>(h0, Wih1, bih1, gi1b, Bb, H3, Hh, stream);
    launch_gemm<1>(h1, Whh1, bhh1, gh1b, Bb, H3, Hh, stream);
    gru_gates_ln_kernel<<<Bb, 256, 0, stream>>>(
        gi1b, (long)H3, gh1b, h1, ln1w, ln1b, h1,
        H1seq + (size_t)t * Hh, (long)Tt * Hh);
  }

  // ---- phase 3 (parallel): head over all 4096 rows ----------------------
  launch_gemm<4>(H1seq, W1, b1, Act, Bb * Tt, Hh, Hh, stream);    // [4096,1024]
  lrelu_ln_kernel<<<Bb * Tt, 256, 0, stream>>>(Act, ln2w, ln2b);
  launch_gemm<4>(Act, W2, b2, out, Bb * Tt, Vv, Hh, stream);      // [4096,32000]
}